// MultiSlotPooling_51986284150863
// MI455X (gfx1250) — compile-verified
//
#include <hip/hip_runtime.h>
#include <cstdint>
#include <cstddef>

typedef __attribute__((ext_vector_type(2))) float v2f;
typedef __attribute__((ext_vector_type(4))) float v4f;
typedef __attribute__((ext_vector_type(8))) float v8f;
typedef int v4i_vs __attribute__((__vector_size__(4 * sizeof(int))));
typedef __attribute__((address_space(1))) v4i_vs* gptr_v4i;
typedef __attribute__((address_space(3))) v4i_vs* lptr_v4i;

#ifndef __has_builtin
#define __has_builtin(x) 0
#endif
#if __has_builtin(__builtin_amdgcn_global_load_async_to_lds_b128) && \
    __has_builtin(__builtin_amdgcn_s_wait_asynccnt)
#define HAVE_ASYNC 1
#else
#define HAVE_ASYNC 0
#endif

#define B_   32
#define T_   4096
#define D_   512
#define K_   8
#define NEGV (-1.0e9f)

#define TPW       128            // t-span per wave (one partial chunk)
#define CH_PER_B  (T_ / TPW)     // 32
#define NCHUNK    (B_ * CH_PER_B)// 1024
#define WAVES_WG  4
#define NTILES    (TPW / 16)     // 8

#define QSTRIDE   516            // 512 + 4 dword pad  (banks: 4*L, conflict-free)
#define HSTRIDE   524            // 512 + 12 dword pad (banks: 12*L, conflict-free both phases)
#define QFLOATS   (16 * QSTRIDE)
#define HTILEF    (16 * HSTRIDE)
#define SMEM_FLOATS (QFLOATS + WAVES_WG * 2 * HTILEF)   // double-buffered tiles
#define SMEM_BYTES  (SMEM_FLOATS * 4)

static __device__ __forceinline__ v8f wmma_f32_16x16x4(v2f a, v2f b, v8f c) {
  return __builtin_amdgcn_wmma_f32_16x16x4_f32(false, a, false, b, (short)0, c,
                                               false, false);
}

// Stage one contiguous 16x512 f32 tile (32 KB) into padded LDS.
// Async path: 64 x global_load_async_to_lds_b128 (ASYNCcnt), no VGPR round trip.
static __device__ __forceinline__ void stage_tile(const float* __restrict__ gsrc,
                                                  float* __restrict__ dst,
                                                  int lane, bool prefetch_next) {
  for (int i = lane * 4; i < 16 * D_; i += 32 * 4) {
    const int r = i >> 9, c = i & (D_ - 1);
#if HAVE_ASYNC
    __builtin_amdgcn_global_load_async_to_lds_b128(
        (gptr_v4i)(gsrc + i),
        (lptr_v4i)(dst + r * HSTRIDE + c),
        0, 0);
    (void)prefetch_next;
#else
    v4f v = *(const v4f*)(gsrc + i);
    *(v4f*)(dst + r * HSTRIDE + c) = v;
    if (prefetch_next) __builtin_prefetch(gsrc + 16 * D_ + i, 0, 1);
#endif
  }
}

static __device__ __forceinline__ void wait_tile() {
#if HAVE_ASYNC
  __builtin_amdgcn_s_wait_asynccnt(0);
#endif
}

// ---------------------------------------------------------------------------
// Kernel 1: per-chunk flash-style pass.  One wave = 128 t of one batch.
//   phase 1: S^T(16t x 16k) = H_tile(16t x 512d) * Q^T(512d x 16k)  [WMMA f32]
//   online softmax per k (k lives in lanes of the D-layout)
//   phase 2: acc(16k x 512d) += P^T(16k x 16t) * H_tile(16t x 512d) [WMMA f32]
// Tiles are double-buffered: tile t+1 streams into LDS (async) while tile t
// is consumed by the WMMA pipeline.
// ---------------------------------------------------------------------------
__launch_bounds__(WAVES_WG * 32, 1)
__global__ void msp_partial_kernel(const float* __restrict__ h,
                                   const unsigned char* __restrict__ mask,
                                   const float* __restrict__ q,
                                   float* __restrict__ ws) {
  extern __shared__ float smem[];
  const int tid  = threadIdx.x;
  const int wave = tid >> 5;
  const int lane = tid & 31;
  const int L    = lane & 15;   // row-within-half / N index
  const int hb   = lane >> 4;   // half-wave select

  // Stage queries once per WG, zero-pad rows 8..15.
  for (int i = tid; i < 16 * D_; i += WAVES_WG * 32) {
    int r = i >> 9, c = i & (D_ - 1);
    smem[r * QSTRIDE + c] = (r < K_) ? q[r * D_ + c] : 0.0f;
  }
  __syncthreads();

  float* ht0 = smem + QFLOATS + (wave * 2 + 0) * HTILEF;
  float* ht1 = smem + QFLOATS + (wave * 2 + 1) * HTILEF;

  const int chunk = blockIdx.x * WAVES_WG + wave;
  const int b  = chunk / CH_PER_B;
  const int t0 = (chunk % CH_PER_B) * TPW;
  const float* gchunk = h + ((size_t)b * T_ + t0) * D_;

  v8f acc[32];                       // acc[nt] : k-rows x 16 d columns
  #pragma unroll
  for (int nt = 0; nt < 32; ++nt)
    #pragma unroll
    for (int j = 0; j < 8; ++j) acc[nt][j] = 0.0f;

  float mrun = -3.0e38f;
  float srun = 0.0f;

  // Prologue: start DMA of tile 0.
  stage_tile(gchunk, ht0, lane, false);

  for (int tile = 0; tile < NTILES; ++tile) {
    const int tbase = t0 + tile * 16;
    float* ht = (tile & 1) ? ht1 : ht0;

    wait_tile();                       // tile `tile` has fully landed in LDS
    if (tile + 1 < NTILES) {           // start DMA of next tile into other buf
      float* nb = ((tile + 1) & 1) ? ht1 : ht0;
      stage_tile(gchunk + (size_t)(tile + 1) * 16 * D_, nb, lane, tile + 2 < NTILES);
    }

    // ---- phase 1: scores (S^T tile) -------------------------------------
    v8f sa;
    #pragma unroll
    for (int j = 0; j < 8; ++j) sa[j] = 0.0f;
    for (int kk = 0; kk < D_; kk += 4) {
      // A-frag (H): lanes0-15 row L K=kk,kk+1 ; lanes16-31 row L K=kk+2,kk+3
      v2f a  = *(const v2f*)(ht   + L * HSTRIDE + kk + 2 * hb);
      // B-frag (Q^T): lane -> k column, same K split
      v2f bq = *(const v2f*)(smem + L * QSTRIDE + kk + 2 * hb);
      sa = wmma_f32_16x16x4(a, bq, sa);
    }

    // ---- mask + online softmax (per column k = lane&15) -----------------
    float sv[8];
    #pragma unroll
    for (int j = 0; j < 8; ++j) {
      int tg = tbase + j + 8 * hb;
      sv[j] = mask[(size_t)b * T_ + tg] ? sa[j] : NEGV;
    }
    float tmax = sv[0];
    #pragma unroll
    for (int j = 1; j < 8; ++j) tmax = fmaxf(tmax, sv[j]);
    tmax = fmaxf(tmax, __shfl_xor(tmax, 16));
    float mnew  = fmaxf(mrun, tmax);
    float scale = __expf(mrun - mnew);
    float pr[8];
    float ls = 0.0f;
    #pragma unroll
    for (int j = 0; j < 8; ++j) { pr[j] = __expf(sv[j] - mnew); ls += pr[j]; }
    ls  += __shfl_xor(ls, 16);
    srun = srun * scale + ls;
    mrun = mnew;

    // Rescale accumulators; acc row k = j + 8*hb, scale for k sits in lane k.
    float rs[8];
    #pragma unroll
    for (int j = 0; j < 8; ++j) rs[j] = __shfl(scale, j + (hb << 3));
    #pragma unroll
    for (int nt = 0; nt < 32; ++nt)
      #pragma unroll
      for (int j = 0; j < 8; ++j) acc[nt][j] *= rs[j];

    // ---- phase 2: acc += P^T * H_tile -----------------------------------
    #pragma unroll
    for (int kt = 0; kt < 16; kt += 4) {
      const int r0 = kt, r1 = kt + 1, r2 = kt + 2, r3 = kt + 3;
      // P[t][k]: component t%8, lane k + 16*(t/8).  Build A-frag rows of P^T.
      float a0lo = __shfl(pr[r0 & 7], L + ((r0 >> 3) << 4));
      float a1lo = __shfl(pr[r1 & 7], L + ((r1 >> 3) << 4));
      float a0hi = __shfl(pr[r2 & 7], L + ((r2 >> 3) << 4));
      float a1hi = __shfl(pr[r3 & 7], L + ((r3 >> 3) << 4));
      v2f a;
      a.x = hb ? a0hi : a0lo;
      a.y = hb ? a1hi : a1lo;
      const int rlo = kt + 2 * hb;      // b-frag K rows for this half
      #pragma unroll
      for (int nt = 0; nt < 32; ++nt) {
        v2f bf;
        bf.x = ht[ rlo      * HSTRIDE + nt * 16 + L];
        bf.y = ht[(rlo + 1) * HSTRIDE + nt * 16 + L];
        acc[nt] = wmma_f32_16x16x4(a, bf, acc[nt]);
      }
    }
  }

  // ---- write partials: [NCHUNK][8] m, [NCHUNK][8] sum, [NCHUNK][8][512] acc
  float* wm   = ws;
  float* wsum = ws + (size_t)NCHUNK * 8;
  float* wacc = ws + (size_t)NCHUNK * 16;
  if (lane < 8) {
    wm  [chunk * 8 + lane] = mrun;   // lane == k for lanes 0..7
    wsum[chunk * 8 + lane] = srun;
  }
  if (lane < 16) {                   // lanes 0-15 hold real rows k = 0..7
    #pragma unroll
    for (int j = 0; j < K_; ++j)
      #pragma unroll
      for (int nt = 0; nt < 32; ++nt)
        wacc[(size_t)chunk * (K_ * D_) + j * D_ + nt * 16 + lane] = acc[nt][j];
  }
}

// ---------------------------------------------------------------------------
// Kernel 2: combine per-chunk partials -> out[b][k][d]
// ---------------------------------------------------------------------------
__global__ void msp_reduce_kernel(const float* __restrict__ ws,
                                  float* __restrict__ out) {
  const int bk = blockIdx.x;
  const int b = bk >> 3, k = bk & 7;
  const int d = threadIdx.x;          // 512 threads
  const float* wm   = ws;
  const float* wsum = ws + (size_t)NCHUNK * 8;
  const float* wacc = ws + (size_t)NCHUNK * 16;
  const int c0 = b * CH_PER_B;

  float M = -3.0e38f;
  for (int c = 0; c < CH_PER_B; ++c) M = fmaxf(M, wm[(c0 + c) * 8 + k]);
  float den = 0.0f, num = 0.0f;
  for (int c = 0; c < CH_PER_B; ++c) {
    float w = __expf(wm[(c0 + c) * 8 + k] - M);
    den += w * wsum[(c0 + c) * 8 + k];
    num += w * wacc[(size_t)(c0 + c) * (K_ * D_) + k * D_ + d];
  }
  out[((size_t)b * K_ + k) * D_ + d] = num / den;
}

extern "C" void kernel_launch(void* const* d_in, const int* in_sizes, int n_in,
                              void* d_out, int out_size, void* d_ws, size_t ws_size,
                              hipStream_t stream) {
  (void)in_sizes; (void)n_in; (void)out_size; (void)ws_size;
  const float*         h    = (const float*)d_in[0];
  const unsigned char* mask = (const unsigned char*)d_in[1];
  const float*         q    = (const float*)d_in[2];
  float* out = (float*)d_out;
  float* ws  = (float*)d_ws;

  (void)hipFuncSetAttribute((const void*)msp_partial_kernel,
                            hipFuncAttributeMaxDynamicSharedMemorySize,
                            SMEM_BYTES);

  msp_partial_kernel<<<NCHUNK / WAVES_WG, WAVES_WG * 32, SMEM_BYTES, stream>>>(
      h, mask, q, ws);
  msp_reduce_kernel<<<B_ * K_, D_, 0, stream>>>(ws, out);
}